// PermutationEntropyModule_81630148428274
// MI455X (gfx1250) — compile-verified
//
#include <hip/hip_runtime.h>
#include <math.h>

#define B_LEN   16
#define S_LEN   8192
#define F_LEN   256
#define NPAT    6
#define WIN     8190                 // S - (m-1)*delay
#define CHUNK_W 126                  // windows per block; 65*126 == 8190 exactly
#define NCHUNK  65
#define ROWS    16                   // s-rows per pipeline stage
#define STAGES  8                    // 8*16 = 128 s-values loaded per chunk
#define ROW_BYTES   (F_LEN * 4)      // 1024
#define STAGE_BYTES (ROWS * ROW_BYTES) // 16384 (global stage stride == LDS buffer size)

// ---- CDNA5 async global->LDS pipeline helpers (per-lane DMA, ASYNCcnt) ----
// INST_OFFSET is added to BOTH the global address and the LDS address
// (CDNA5 ISA); LDS row stride == global row stride (1024 B),
// so one VGPR address pair + one LDS base register serve a whole stage.
__device__ __forceinline__ void async_stage_16(unsigned lds_byte, unsigned long long gaddr) {
#pragma unroll
  for (int r = 0; r < ROWS; ++r) {
    asm volatile("global_load_async_to_lds_b32 %0, %1, off offset:%c2"
                 :: "v"(lds_byte), "v"(gaddr), "i"(r * ROW_BYTES)
                 : "memory");
  }
}
__device__ __forceinline__ void wait_async_le16() {
  asm volatile("s_wait_asynccnt 0x10" ::: "memory");
}
__device__ __forceinline__ void wait_async_0() {
  asm volatile("s_wait_asynccnt 0x0" ::: "memory");
}

// hardware f32 atomic add (global_atomic_add_f32), no CAS loop
__device__ __forceinline__ void hw_atomic_add_f32(float* p, float v) {
  unsafeAtomicAdd(p, v);
}

// One row of the sliding-window pipeline. `window`/`ownE` fold to constants
// at every call site. q2,q1 carry the squares of p2,p1 (computed once/value).
__device__ __forceinline__ void proc_row(float v, bool window, bool ownE,
                                         float& p2, float& p1,
                                         float& q2, float& q1,
                                         float& e,
                                         float& h0, float& h1, float& h2,
                                         float& h3, float& h4, float& h5)
{
  const float q0 = v * v;
  if (ownE) e += q0;
  if (window) {
    const float a = p2, bb = p1, cc = v;
    // stable-argsort Lehmer rank via 3 comparisons + LUT
    const bool ab = a  <= bb;
    const bool ac = a  <= cc;
    const bool bc = bb <= cc;
    const int  key = (ab ? 4 : 0) | (ac ? 2 : 0) | (bc ? 1 : 0);
    const int  rk  = (int)((0x01042035u >> (key * 4)) & 7u);
    // amplitude weight: var*(1+0.5*range)*(l2/m)
    const float mu  = (a + bb + cc) * (1.0f / 3.0f);
    const float d0 = a - mu, d1 = bb - mu, d2 = cc - mu;
    const float var = (d0 * d0 + d1 * d1 + d2 * d2) * (1.0f / 3.0f) + 1e-10f;
    const float mx  = fmaxf(a, fmaxf(bb, cc));
    const float mn  = fminf(a, fminf(bb, cc));
    const float rng = mx - mn;
    const float l2  = sqrtf(q2 + q1 + q0) * (1.0f / 3.0f);
    const float w   = var * (1.0f + 0.5f * rng) * l2;
    h0 += (rk == 0) ? w : 0.0f;
    h1 += (rk == 1) ? w : 0.0f;
    h2 += (rk == 2) ? w : 0.0f;
    h3 += (rk == 3) ? w : 0.0f;
    h4 += (rk == 4) ? w : 0.0f;
    h5 += (rk == 5) ? w : 0.0f;
  }
  p2 = p1; p1 = v;
  q2 = q1; q1 = q0;
}

// One block = (batch b, chunk c of 126 windows). Thread = one f column.
// Streams 128 s-rows through a double-buffered async-LDS pipeline:
// stage 0 peeled (warm-up), stages 1..6 rolled & guard-free, stage 7 peeled.
__global__ void __launch_bounds__(F_LEN)
aape_stream_kernel(const float* __restrict__ x,
                   float* __restrict__ hist,    // [B][F][NPAT]
                   float* __restrict__ energy)  // [B][F]
{
  __shared__ float smem[2][ROWS][F_LEN];
  const int f  = threadIdx.x;
  const int c  = blockIdx.x;
  const int b  = blockIdx.y;
  const int s0 = c * CHUNK_W;
  const bool lastChunk = (c == NCHUNK - 1);  // also owns energy rows 126,127

  const unsigned long long g0 =
      (unsigned long long)(x + (((size_t)b * S_LEN + (size_t)s0) * F_LEN + (size_t)f));
  // low 32 bits of a generic LDS pointer == LDS byte offset
  const unsigned lds0 = (unsigned)(unsigned long long)(void*)&smem[0][0][f];

  // prime two stages
  async_stage_16(lds0,               g0);
  async_stage_16(lds0 + STAGE_BYTES, g0 + STAGE_BYTES);

  float h0 = 0.f, h1 = 0.f, h2 = 0.f, h3 = 0.f, h4 = 0.f, h5 = 0.f;
  float e  = 0.f;
  float p2 = 0.f, p1 = 0.f;   // sliding window: values at s-2, s-1
  float q2 = 0.f, q1 = 0.f;   // their squares

  // ---- stage 0 (buf 0): rows 0,1 have no window yet ----
  wait_async_le16();
#pragma unroll
  for (int r = 0; r < ROWS; ++r)
    proc_row(smem[0][r][f], r >= 2, true, p2, p1, q2, q1, e, h0, h1, h2, h3, h4, h5);
  async_stage_16(lds0, g0 + 2ull * STAGE_BYTES);   // stage 2 -> buf 0

  // ---- stages 1..6: steady state, no guards ----
#pragma unroll 1
  for (int st = 1; st <= 6; ++st) {
    wait_async_le16();                             // stage st complete (in-order)
    const int buf = st & 1;
#pragma unroll
    for (int r = 0; r < ROWS; ++r)
      proc_row(smem[buf][r][f], true, true, p2, p1, q2, q1, e, h0, h1, h2, h3, h4, h5);
    if (st < 6)
      async_stage_16(lds0 + (unsigned)buf * STAGE_BYTES,
                     g0 + (unsigned long long)(st + 2) * STAGE_BYTES);
  }

  // ---- stage 7 (buf 1): energy tail rows 126,127 owned only by last chunk ----
  wait_async_0();
#pragma unroll
  for (int r = 0; r < ROWS; ++r)
    proc_row(smem[1][r][f], true, (r < 14) || lastChunk,
             p2, p1, q2, q1, e, h0, h1, h2, h3, h4, h5);

  float* hp = hist + (size_t)(b * F_LEN + f) * NPAT;
  hw_atomic_add_f32(hp + 0, h0);
  hw_atomic_add_f32(hp + 1, h1);
  hw_atomic_add_f32(hp + 2, h2);
  hw_atomic_add_f32(hp + 3, h3);
  hw_atomic_add_f32(hp + 4, h4);
  hw_atomic_add_f32(hp + 5, h5);
  hw_atomic_add_f32(energy + b * F_LEN + f, e);
}

__global__ void __launch_bounds__(256)
zero_ws_kernel(float* __restrict__ p, int n) {
  int i = blockIdx.x * blockDim.x + threadIdx.x;
  if (i < n) p[i] = 0.0f;
}

// blocks 0..15: energy_ratio rows; block 16: entropy -> scaled scores
__global__ void __launch_bounds__(F_LEN)
finalize_kernel(const float* __restrict__ sens,
                const float* __restrict__ hist,
                const float* __restrict__ energy,
                float* __restrict__ out)
{
  const int f = threadIdx.x;
  if (blockIdx.x < B_LEN) {
    const int b = blockIdx.x;
    __shared__ float red[F_LEN];
    const float em = energy[b * F_LEN + f] * (1.0f / (float)S_LEN);
    red[f] = em;
    __syncthreads();
    for (int off = F_LEN / 2; off > 0; off >>= 1) {
      if (f < off) red[f] += red[f + off];
      __syncthreads();
    }
    const float total = red[0];
    out[F_LEN + b * F_LEN + f] = em / (total + 1e-8f);
  } else {
    float acc = 0.0f;
    for (int b = 0; b < B_LEN; ++b) {
      const float* h = hist + (size_t)(b * F_LEN + f) * NPAT;
      float hv[NPAT];
      float t = 0.0f;
#pragma unroll
      for (int k = 0; k < NPAT; ++k) { hv[k] = h[k]; t += hv[k]; }
      float ent = 0.0f;
      if (t > 0.0f) {
        const float inv = 1.0f / t;
#pragma unroll
        for (int k = 0; k < NPAT; ++k) {
          if (hv[k] > 0.0f) { const float p = hv[k] * inv; ent -= p * log2f(p); }
        }
      }
      acc += ent;
    }
    const float aape = acc * (1.0f / (float)B_LEN) * (1.0f / 2.5849625007211562f);
    const float sg = 1.0f / (1.0f + expf(-sens[0]));
    out[f] = aape * sg;
  }
}

extern "C" void kernel_launch(void* const* d_in, const int* in_sizes, int n_in,
                              void* d_out, int out_size, void* d_ws, size_t ws_size,
                              hipStream_t stream) {
  (void)in_sizes; (void)n_in; (void)out_size; (void)ws_size;
  const float* x    = (const float*)d_in[0];
  const float* sens = (const float*)d_in[1];
  float* out  = (float*)d_out;
  float* hist = (float*)d_ws;                       // 16*256*6 floats
  float* en   = hist + B_LEN * F_LEN * NPAT;        // 16*256 floats

  const int nz = B_LEN * F_LEN * (NPAT + 1);        // 28672 floats
  zero_ws_kernel<<<(nz + 255) / 256, 256, 0, stream>>>((float*)d_ws, nz);
  aape_stream_kernel<<<dim3(NCHUNK, B_LEN), dim3(F_LEN), 0, stream>>>(x, hist, en);
  finalize_kernel<<<dim3(B_LEN + 1), dim3(F_LEN), 0, stream>>>(sens, hist, en, out);
}